// ResidualVectorQuantizer_523986010686
// MI455X (gfx1250) — compile-verified
//
#include <hip/hip_runtime.h>
#include <hip/hip_bf16.h>

// ---------------------------------------------------------------------------
// Residual Vector Quantizer for MI455X (gfx1250, wave32, WMMA f32 16x16x4)
//
// x[32][256][2048] f32 -> quantized[32][256][2048], codes[8][32][2048],
// bw[1], commit_loss[8] (all f32, concatenated in d_out).
//
// One wave owns 32 tokens (two 16-row WMMA M-tiles sharing every B fragment);
// one block = 2 waves = 64 tokens. Residual tile lives in dynamic LDS; all
// three per-stage GEMMs run on v_wmma_f32_16x16x4_f32 with biases / -0.5|cb|^2
// folded into the WMMA C operand. quantized = x - residual_final (no accum).
// ---------------------------------------------------------------------------

typedef float v2f __attribute__((ext_vector_type(2)));
typedef float v8f __attribute__((ext_vector_type(8)));

#define NQ      8
#define BINS    1024
#define DIMD    256
#define CBD     8
#define TOKS    65536            // B*T = 32*2048
#define TPB     64               // tokens per block (2 waves x 32)
#define PITCH   258              // residual row pitch in floats (even, bank-spread)

#define Q_OFF   0
#define C_OFF   16777216         // 32*256*2048
#define BW_OFF  17301504         // C_OFF + 8*32*2048
#define CL_OFF  17301505

// Builtin prototype (from hipcc diagnostic): global operand is a non-const
// address_space(1) int*; LDS operand is the address_space(3) counterpart.
#define GLB_PTR(p) ((__attribute__((address_space(1))) int*)(p))
#define LDS_PTR(p) ((__attribute__((address_space(3))) int*)(p))

static __device__ __forceinline__ v8f wmma4(v2f a, v2f b, v8f c) {
  // D = A(16x4 f32) * B(4x16 f32) + C(16x16 f32)
  return __builtin_amdgcn_wmma_f32_16x16x4_f32(
      /*neg_a=*/false, a, /*neg_b=*/false, b,
      /*c_mod=*/(short)0, c, /*reuse_a=*/false, /*reuse_b=*/false);
}

static __device__ __forceinline__ void wait_async0() {
#if __has_builtin(__builtin_amdgcn_s_wait_asynccnt)
  __builtin_amdgcn_s_wait_asynccnt(0);
#else
  asm volatile("s_wait_asynccnt 0x0" ::: "memory");
#endif
}

// ---------------------------------------------------------------------------
// Init: codebook norms (-0.5*|cb|^2) into workspace; bw + zeroed commit slots.
// ---------------------------------------------------------------------------
__global__ __launch_bounds__(256) void rvq_init(const float* __restrict__ cbs,
                                                const int* __restrict__ fr,
                                                float* __restrict__ norms,
                                                float* __restrict__ out) {
  int i = blockIdx.x * blockDim.x + threadIdx.x;
  if (i < NQ * BINS) {
    const float* c = cbs + (size_t)i * CBD;
    float s = 0.f;
#pragma unroll
    for (int j = 0; j < CBD; ++j) s += c[j] * c[j];
    norms[i] = -0.5f * s;
  }
  if (i == 0) out[BW_OFF] = (float)(NQ * 10 * fr[0]);  // log2(1024) == 10
  if (i < NQ) out[CL_OFF + i] = 0.f;
}

// ---------------------------------------------------------------------------
// Main kernel.
// ---------------------------------------------------------------------------
__global__ __launch_bounds__(64) void rvq_main(const float* __restrict__ x,
                                               const float* __restrict__ Win,
                                               const float* __restrict__ bin,
                                               const float* __restrict__ cbs,
                                               const float* __restrict__ Wout,
                                               const float* __restrict__ bout,
                                               const float* __restrict__ norms,
                                               float* __restrict__ out) {
  extern __shared__ float s_res[];            // [64][PITCH] residual tile (dynamic)
  __shared__ float s_h[2][32][CBD];           // per-wave h scratch
  __shared__ float s_q[2][32][CBD];           // per-wave quant scratch
  __shared__ float s_redv[2][32][16];         // argmax reduce: values
  __shared__ int   s_redi[2][32][16];         // argmax reduce: indices
  __shared__ float s_commit[NQ];

  const int tid  = threadIdx.x;
  const int wave = tid >> 5;
  const int lane = tid & 31;
  const int col  = lane & 15;                 // N / column slot
  const int hi   = lane >> 4;                 // half-wave (K / M split)
  const int rb   = wave * 32;                 // this wave's local token base

  const int g0 = blockIdx.x * TPB;            // global token base
  const int b  = g0 >> 11;                    // g0 / 2048
  const int t0 = g0 & 2047;

  if (tid < NQ) s_commit[tid] = 0.f;

  // ---- load residual tile: res[tl][d] = x[b][d][t0+tl] -------------------
  // Coalesced over t across lanes; async copy straight into LDS if available.
  {
    const int tl = tid;                       // 64 threads = 64 tokens
    __builtin_prefetch(x + ((size_t)b * DIMD) * 2048 + t0, 0, 1);
#if __has_builtin(__builtin_amdgcn_global_load_async_to_lds_b32)
    for (int d = 0; d < DIMD; ++d) {
      __builtin_amdgcn_global_load_async_to_lds_b32(
          GLB_PTR(x + ((size_t)b * DIMD + d) * 2048 + t0 + tl),
          LDS_PTR(s_res + tl * PITCH + d), /*offset=*/0, /*cpol=*/0);
    }
    wait_async0();
#else
    for (int d = 0; d < DIMD; ++d) {
      s_res[tl * PITCH + d] = x[((size_t)b * DIMD + d) * 2048 + t0 + tl];
    }
#endif
  }
  __syncthreads();

  // ---- 8 sequential stages, wave-private (no inter-wave barriers) ---------
  for (int q = 0; q < NQ; ++q) {
    const float* Wq  = Win  + (size_t)q * CBD * DIMD;   // [8][256]
    const float* cbq = cbs  + (size_t)q * BINS * CBD;   // [1024][8]
    const float* Woq = Wout + (size_t)q * DIMD * CBD;   // [256][8]
    const float* nq  = norms + (size_t)q * BINS;

    // ===== (1) h = residual * W_in^T + b_in  (two M-tiles share B) =========
    v8f acc0, acc1;
    {
      float binv = 0.f;
      if (col < CBD) binv = bin[q * CBD + col];
#pragma unroll
      for (int r = 0; r < 8; ++r) { acc0[r] = binv; acc1[r] = binv; }
    }
    const float vmask = (col < CBD) ? 1.f : 0.f;
#pragma unroll 4
    for (int kc = 0; kc < 64; ++kc) {
      // B fragment: (K=d, N=c) = W_in[c][d]; per-lane K-pair contiguous in d
      v2f bf = *(const v2f*)(Wq + (size_t)(col & 7) * DIMD + kc * 4 + hi * 2);
      bf = bf * vmask;                        // zero the 8 padded columns
      // A fragments: row M=col (+16), K = 4*kc + 2*hi + {0,1}
      v2f a0 = *(const v2f*)(&s_res[(rb + col) * PITCH + kc * 4 + hi * 2]);
      v2f a1 = *(const v2f*)(&s_res[(rb + 16 + col) * PITCH + kc * 4 + hi * 2]);
      acc0 = wmma4(a0, bf, acc0);
      acc1 = wmma4(a1, bf, acc1);
    }
    // spill h (D-layout) -> row-major scratch for re-fragmentation
    if (col < CBD) {
#pragma unroll
      for (int r = 0; r < 8; ++r) {
        s_h[wave][r + 8 * hi][col]      = acc0[r];
        s_h[wave][16 + r + 8 * hi][col] = acc1[r];
      }
    }

    // A fragments of H for the distance GEMM (K = CBD = 8 -> two k-chunks)
    v2f HA0a = *(const v2f*)(&s_h[wave][col][2 * hi]);
    v2f HA1a = *(const v2f*)(&s_h[wave][col][4 + 2 * hi]);
    v2f HA0b = *(const v2f*)(&s_h[wave][16 + col][2 * hi]);
    v2f HA1b = *(const v2f*)(&s_h[wave][16 + col][4 + 2 * hi]);

    // ===== (2) score = h . cb_n - 0.5|cb_n|^2 ; running argmax over 1024 ===
    float bestv0[8], bestv1[8];
    int   besti0[8], besti1[8];
#pragma unroll
    for (int r = 0; r < 8; ++r) {
      bestv0[r] = -3.0e38f; besti0[r] = 0;
      bestv1[r] = -3.0e38f; besti1[r] = 0;
    }
#pragma unroll 2
    for (int nt = 0; nt < BINS / 16; ++nt) {
      const int n = nt * 16 + col;
      const float nrm = nq[n];                // -0.5*|cb_n|^2 folded via C
      const float* cbp = cbq + (size_t)n * CBD + 2 * hi;
      v2f B0 = *(const v2f*)(cbp);            // K = 2hi + {0,1}
      v2f B1 = *(const v2f*)(cbp + 4);        // K = 4 + 2hi + {0,1}
      v8f c0, c1;
#pragma unroll
      for (int r = 0; r < 8; ++r) { c0[r] = nrm; c1[r] = nrm; }
      c0 = wmma4(HA0a, B0, c0);
      c0 = wmma4(HA1a, B1, c0);
      c1 = wmma4(HA0b, B0, c1);               // second M-tile reuses B
      c1 = wmma4(HA1b, B1, c1);
#pragma unroll
      for (int r = 0; r < 8; ++r) {
        if (c0[r] > bestv0[r]) { bestv0[r] = c0[r]; besti0[r] = nt; }
        if (c1[r] > bestv1[r]) { bestv1[r] = c1[r]; besti1[r] = nt; }
      }
    }
    // cross-lane argmax: token M's 16 candidates live across the 16 columns
#pragma unroll
    for (int r = 0; r < 8; ++r) {
      const int tok = r + 8 * hi;
      s_redv[wave][tok][col]      = bestv0[r];
      s_redi[wave][tok][col]      = besti0[r] * 16 + col;
      s_redv[wave][16 + tok][col] = bestv1[r];
      s_redi[wave][16 + tok][col] = besti1[r] * 16 + col;
    }
    {
      const int tok = lane;                   // 32 lanes, 32 tokens: full wave
      float bv = s_redv[wave][tok][0];
      int   bi = s_redi[wave][tok][0];
#pragma unroll
      for (int j = 1; j < 16; ++j) {
        float v = s_redv[wave][tok][j];
        int   i = s_redi[wave][tok][j];
        if (v > bv) { bv = v; bi = i; }
      }
      const int gt = g0 + rb + tok;
      out[C_OFF + (size_t)q * TOKS + gt] = (float)bi;  // codes as f32
      // quant lookup + commit partial
      const float* cv = cbq + (size_t)bi * CBD;
      float cm = 0.f;
#pragma unroll
      for (int c2 = 0; c2 < CBD; ++c2) {
        float qv = cv[c2];
        s_q[wave][tok][c2] = qv;
        float dd = qv - s_h[wave][tok][c2];
        cm += dd * dd;
      }
      atomicAdd(&s_commit[q], cm);
    }

    // A fragments of quant for the output projection
    v2f QA0a = *(const v2f*)(&s_q[wave][col][2 * hi]);
    v2f QA1a = *(const v2f*)(&s_q[wave][col][4 + 2 * hi]);
    v2f QA0b = *(const v2f*)(&s_q[wave][16 + col][2 * hi]);
    v2f QA1b = *(const v2f*)(&s_q[wave][16 + col][4 + 2 * hi]);

    // ===== (3) out = quant * W_out^T + b_out ; residual -= out =============
#pragma unroll 2
    for (int nt = 0; nt < DIMD / 16; ++nt) {
      const int d = nt * 16 + col;
      const float bo = bout[q * DIMD + d];
      const float* wp = Woq + (size_t)d * CBD + 2 * hi;
      v2f B0 = *(const v2f*)(wp);
      v2f B1 = *(const v2f*)(wp + 4);
      v8f c0, c1;
#pragma unroll
      for (int r = 0; r < 8; ++r) { c0[r] = bo; c1[r] = bo; }
      c0 = wmma4(QA0a, B0, c0);
      c0 = wmma4(QA1a, B1, c0);
      c1 = wmma4(QA0b, B0, c1);               // second M-tile reuses B
      c1 = wmma4(QA1b, B1, c1);
#pragma unroll
      for (int r = 0; r < 8; ++r) {
        const int tok = r + 8 * hi;
        s_res[(rb + tok) * PITCH + d]      -= c0[r];
        s_res[(rb + 16 + tok) * PITCH + d] -= c1[r];
      }
    }
  }

  __syncthreads();
  if (tid < NQ) {
    atomicAdd(&out[CL_OFF + tid], s_commit[tid] * (1.0f / (float)(TOKS * CBD)));
  }

  // ---- quantized = x - residual_final (transpose-free: same layout as x) --
  {
    const int tl = tid;
    for (int d = 0; d < DIMD; ++d) {
      const size_t xi = ((size_t)b * DIMD + d) * 2048 + t0 + tl;
      out[Q_OFF + xi] = x[xi] - s_res[tl * PITCH + d];
    }
  }
}

// ---------------------------------------------------------------------------
extern "C" void kernel_launch(void* const* d_in, const int* in_sizes, int n_in,
                              void* d_out, int out_size, void* d_ws, size_t ws_size,
                              hipStream_t stream) {
  (void)in_sizes; (void)n_in; (void)out_size; (void)ws_size;
  const float* x    = (const float*)d_in[0];
  const float* Win  = (const float*)d_in[1];
  const float* bin  = (const float*)d_in[2];
  const float* cbs  = (const float*)d_in[3];
  const float* Wout = (const float*)d_in[4];
  const float* bout = (const float*)d_in[5];
  const int*   fr   = (const int*)d_in[6];
  float* out   = (float*)d_out;
  float* norms = (float*)d_ws;               // 8*1024 floats = 32 KB scratch

  const size_t dyn_lds = (size_t)TPB * PITCH * sizeof(float);  // 66,048 B
  rvq_init<<<(NQ * BINS + 255) / 256, 256, 0, stream>>>(cbs, fr, norms, out);
  rvq_main<<<TOKS / TPB, 64, dyn_lds, stream>>>(x, Win, bin, cbs, Wout, bout,
                                                norms, out);
}